// CraftLoss_38293928411859
// MI455X (gfx1250) — compile-verified
//
#include <hip/hip_runtime.h>

// ---------------------------------------------------------------------------
// CRAFT loss on MI455X (gfx1250): pure streaming reduction, HBM-bound.
// 151 MB in / 4 B out -> ~6.5 us floor at 23.3 TB/s. Strategy:
//   * b128 non-temporal loads (single-use stream, TH=NT)
//   * wave32 shfl reductions; 8 waves x 4 accumulators == one wave32
//   * deterministic 2-pass reduction through d_ws (no float atomics)
//   * final 32-lane sum done with V_WMMA_F32_16X16X4_F32 (A = data in
//     K0/K2 slots per ISA 16x4 f32 layout, B = ones -> row sums)
// ---------------------------------------------------------------------------

typedef float vf4 __attribute__((ext_vector_type(4)));
typedef float v2f __attribute__((ext_vector_type(2)));
typedef float v8f __attribute__((ext_vector_type(8)));

#define POS_THRES 0.1f
#define NEG_THRES 0.0f
#define NPIX      (16 * 768 * 768)   /* 9,437,184 pixels               */
#define NGROUPS   (NPIX / 4)         /* 2,359,296 float4 pixel groups  */
#define NBLOCKS   2048
#define NTHREADS  256

__device__ __forceinline__ void accum_pixel(float pred, float tgt,
                                            float& num, float& cnt) {
  float d = pred - tgt;
  float l = d * d;
  bool pos = tgt >= POS_THRES;
  bool neg = tgt <= NEG_THRES;
  num += pos ? l : 0.0f;   // positive_loss contribution
  num += neg ? l : 0.0f;   // negative_loss contribution (mutually exclusive)
  cnt += pos ? 1.0f : 0.0f;
}

// ---------------------------------------------------------------------------
// Pass 1: grid-stride streaming + per-block reduction -> 4 partials per block
// ---------------------------------------------------------------------------
__global__ __launch_bounds__(NTHREADS) void craft_partial_kernel(
    const float* __restrict__ outp,   // (B,H,W,2) interleaved char/aff preds
    const float* __restrict__ cmap,   // (B,H,W)
    const float* __restrict__ amap,   // (B,H,W)
    float* __restrict__ partials)     // [NBLOCKS][4]
{
  const vf4* o4 = (const vf4*)outp;
  const vf4* c4 = (const vf4*)cmap;
  const vf4* a4 = (const vf4*)amap;

  float cnum = 0.0f, cpos = 0.0f, anum = 0.0f, apos = 0.0f;

  const unsigned stride = gridDim.x * blockDim.x;
  for (unsigned g = blockIdx.x * blockDim.x + threadIdx.x; g < NGROUPS;
       g += stride) {
    // 64 bytes per iteration, all b128, all non-temporal (single-use stream)
    vf4 ct = __builtin_nontemporal_load(&c4[g]);
    vf4 at = __builtin_nontemporal_load(&a4[g]);
    vf4 o0 = __builtin_nontemporal_load(&o4[2u * g]);      // c0 a0 c1 a1
    vf4 o1 = __builtin_nontemporal_load(&o4[2u * g + 1u]); // c2 a2 c3 a3

    accum_pixel(o0.x, ct.x, cnum, cpos);
    accum_pixel(o0.z, ct.y, cnum, cpos);
    accum_pixel(o1.x, ct.z, cnum, cpos);
    accum_pixel(o1.z, ct.w, cnum, cpos);

    accum_pixel(o0.y, at.x, anum, apos);
    accum_pixel(o0.w, at.y, anum, apos);
    accum_pixel(o1.y, at.z, anum, apos);
    accum_pixel(o1.w, at.w, anum, apos);
  }

  // wave32 intra-wave reduction
  #pragma unroll
  for (int off = 16; off > 0; off >>= 1) {
    cnum += __shfl_xor(cnum, off, 32);
    cpos += __shfl_xor(cpos, off, 32);
    anum += __shfl_xor(anum, off, 32);
    apos += __shfl_xor(apos, off, 32);
  }

  // 8 waves x 4 accumulators == exactly 32 values -> one wave32 finishes it
  __shared__ float red[8][4];
  const int lane = threadIdx.x & 31;
  const int wid  = threadIdx.x >> 5;
  if (lane == 0) {
    red[wid][0] = cnum; red[wid][1] = cpos;
    red[wid][2] = anum; red[wid][3] = apos;
  }
  __syncthreads();
  if (wid == 0) {
    // lanes 0-7: acc0 over 8 waves; 8-15: acc1; 16-23: acc2; 24-31: acc3
    float v = red[lane & 7][lane >> 3];
    #pragma unroll
    for (int off = 4; off > 0; off >>= 1)
      v += __shfl_xor(v, off, 32);     // xor stays inside each 8-lane group
    if ((lane & 7) == 0)
      partials[blockIdx.x * 4 + (lane >> 3)] = v;
  }
}

// ---------------------------------------------------------------------------
// Pass 2: 4 waves, one accumulator each; WMMA does the 32-lane reduction.
// ---------------------------------------------------------------------------
__global__ __launch_bounds__(128) void craft_final_kernel(
    const float* __restrict__ partials,  // [NBLOCKS][4]
    float* __restrict__ out)
{
  const int lane = threadIdx.x & 31;
  const int wid  = threadIdx.x >> 5;    // accumulator id 0..3

  float s = 0.0f;
  #pragma unroll 4
  for (int b = lane; b < NBLOCKS; b += 32)   // uniform trip count -> EXEC all 1s
    s += partials[b * 4 + wid];

  // V_WMMA_F32_16X16X4_F32 as a wave32 lane-reduction:
  // A 16x4 layout (ISA 7.12.2): lanes 0-15 {v0:K0, v1:K1}, lanes 16-31 {v0:K2, v1:K3}
  // a = {s, 0} => A[m][0]=s_m, A[m][2]=s_{m+16}, A[m][1]=A[m][3]=0
  // B = ones(4x16) (layout-independent) => D[m][n] = s_m + s_{m+16}
  v2f a;  a.x = s;    a.y = 0.0f;
  v2f bb; bb.x = 1.0f; bb.y = 1.0f;
  v8f c = {};
  c = __builtin_amdgcn_wmma_f32_16x16x4_f32(
      /*neg_a=*/false, a, /*neg_b=*/false, bb,
      /*c_mod=*/(short)0, c, /*reuse_a=*/false, /*reuse_b=*/false);

  // Lane L (L<16) holds D[0..7][L]; lane L+16 holds D[8..15][L].
  float t = c[0] + c[1] + c[2] + c[3] + c[4] + c[5] + c[6] + c[7];
  float tot = t + __shfl(t, 16, 32);   // lane 0: rows 0-7 + rows 8-15 = full sum

  __shared__ float res[4];
  if (lane == 0) res[wid] = tot;
  __syncthreads();

  if (threadIdx.x == 0) {
    const float n = (float)NPIX;
    float loss_char = res[0] / (res[1] + n);   // (pos+neg loss) / (pos_cnt + n)
    float loss_aff  = res[2] / (res[3] + n);
    out[0] = (loss_char * 2.0f + loss_aff) * 100.0f;
  }
}

// ---------------------------------------------------------------------------
extern "C" void kernel_launch(void* const* d_in, const int* in_sizes, int n_in,
                              void* d_out, int out_size, void* d_ws, size_t ws_size,
                              hipStream_t stream) {
  const float* outp = (const float*)d_in[0];   // output (B,H,W,2)
  const float* cmap = (const float*)d_in[1];   // character_map
  const float* amap = (const float*)d_in[2];   // affinity_map
  float* partials   = (float*)d_ws;            // NBLOCKS*4 floats = 32 KB

  craft_partial_kernel<<<NBLOCKS, NTHREADS, 0, stream>>>(outp, cmap, amap,
                                                         partials);
  craft_final_kernel<<<1, 128, 0, stream>>>(partials, (float*)d_out);
}